// BandDecimate_55113020342683
// MI455X (gfx1250) — compile-verified
//
#include <hip/hip_runtime.h>
#include <cmath>
#include <complex>

#ifndef M_PI
#define M_PI 3.14159265358979323846
#endif

// ---------------- problem geometry ----------------
#define TIN   65536            // samples per channel
#define EDGE  27               // filtfilt odd-extension length (3*9)
#define TEXT  (TIN + 2*EDGE)   // 65590 extended length
#define LP    1032             // partition length (129 blocks of 8)
#define NPART 64
#define TP    (LP * NPART)     // 66048 padded extended length
#define NCH   512              // 32*16 channels
#define NBLK  (LP / 8)         // 129 blocks per partition
#define QDEC  4

// ---------------- workspace layout (float offsets) ----------------
#define OFF_M   ((size_t)0)                        // 16x16 block matrix (256)
#define OFF_R   ((size_t)256)                      // 8x8  A^LP          (64)
#define OFF_ZI  ((size_t)320)                      // zi                 (8)
#define OFF_G   ((size_t)512)                      // LP x 8 zero-input response (8256)
#define OFF_Y   ((size_t)16384)                    // time-major forward output TP*NCH
#define OFF_S   (OFF_Y + (size_t)TP * NCH)         // NPART*NCH*8 exit states
#define OFF_ZIN (OFF_S + (size_t)NPART * NCH * 8)  // NPART*NCH*8 entry states

typedef __attribute__((ext_vector_type(2))) float v2f;
typedef __attribute__((ext_vector_type(8))) float v8f;

struct Coef { double b[9]; double a[9]; double zi[8]; };

// ---------------- fp32 WMMA 16x16x4 wrapper ----------------
__device__ inline v8f wmma4(v2f a, v2f b, v8f c) {
  // (neg_a, A, neg_b, B, c_mod, C, reuse_a, reuse_b)
  return __builtin_amdgcn_wmma_f32_16x16x4_f32(false, a, false, b, (short)0, c,
                                               false, false);
}

// odd-extended input, channel-major source, "ext time" te in [0, TP)  (slow path)
__device__ inline float load_ext(const float* __restrict__ x, int ch, int te) {
  const float* row = x + (size_t)ch * TIN;
  int t = te - EDGE;
  if (te < EDGE)  return 2.0f * row[0] - row[EDGE - te];
  if (t < TIN)    return row[t];
  int j = t - TIN;
  if (j < EDGE)   return 2.0f * row[TIN - 1] - row[TIN - 2 - j];
  return 0.0f;   // padding past TEXT
}

// reversed forward output (time-major), backward time tb  (slow path)
__device__ inline float load_rev(const float* __restrict__ yb, int ch, int tb) {
  int tf = TEXT - 1 - tb;
  return (tf >= 0) ? yb[(size_t)tf * NCH + ch] : 0.0f;
}

// ---------------- precompute M, G, R, zi on device (double) ----------------
__global__ __launch_bounds__(32) void kPrecomp(Coef cf, float* __restrict__ ws) {
  int tid = threadIdx.x;
  if (tid < 16) {
    // column tid of M: unit x_j (tid<8) or unit z_i (tid>=8), run 8 DF2T steps
    double z[8]; for (int i = 0; i < 8; ++i) z[i] = 0.0;
    if (tid >= 8) z[tid - 8] = 1.0;
    double ycol[8];
    for (int t = 0; t < 8; ++t) {
      double xt = (tid < 8 && t == tid) ? 1.0 : 0.0;
      double yt = cf.b[0] * xt + z[0];
      double zn[8];
      for (int i = 0; i < 7; ++i) zn[i] = z[i + 1] + cf.b[i + 1] * xt - cf.a[i + 1] * yt;
      zn[7] = cf.b[8] * xt - cf.a[8] * yt;
      for (int i = 0; i < 8; ++i) z[i] = zn[i];
      ycol[t] = yt;
    }
    for (int r = 0; r < 8; ++r) {
      ws[OFF_M + (size_t)r * 16 + tid]       = (float)ycol[r];
      ws[OFF_M + (size_t)(8 + r) * 16 + tid] = (float)z[r];
    }
  }
  if (tid < 8) {
    // zero-input response column tid: G[t][tid] = (A^t e_tid)[0]; R = A^LP
    double z[8]; for (int i = 0; i < 8; ++i) z[i] = 0.0;
    z[tid] = 1.0;
    for (int t = 0; t < LP; ++t) {
      ws[OFF_G + (size_t)t * 8 + tid] = (float)z[0];
      double yt = z[0];
      double zn[8];
      for (int i = 0; i < 7; ++i) zn[i] = z[i + 1] - cf.a[i + 1] * yt;
      zn[7] = -cf.a[8] * yt;
      for (int i = 0; i < 8; ++i) z[i] = zn[i];
    }
    for (int r = 0; r < 8; ++r) ws[OFF_R + (size_t)r * 8 + tid] = (float)z[r];
  }
  if (tid == 0)
    for (int i = 0; i < 8; ++i) ws[OFF_ZI + i] = (float)cf.zi[i];
}

// ---------------- Phase A forward: zero-state block-WMMA scan ----------------
__global__ __launch_bounds__(32) void kPhaseAFwd(const float* __restrict__ x,
                                                 float* __restrict__ ws) {
  const int lane = threadIdx.x;
  const int p  = blockIdx.x >> 5;     // partition
  const int cg = blockIdx.x & 31;     // channel group of 16
  const int c  = lane & 15;
  const int ch = cg * 16 + c;
  float* yb = ws + OFF_Y;
  float* sb = ws + OFF_S;
  const float* Mf = ws + OFF_M;
  const int coff = (lane >> 4) * 2;   // lane-half K split of A operand
  v2f A0, A1, A2, A3;
  A0[0] = Mf[c * 16 +  0 + coff]; A0[1] = Mf[c * 16 +  1 + coff];
  A1[0] = Mf[c * 16 +  4 + coff]; A1[1] = Mf[c * 16 +  5 + coff];
  A2[0] = Mf[c * 16 +  8 + coff]; A2[1] = Mf[c * 16 +  9 + coff];
  A3[0] = Mf[c * 16 + 12 + coff]; A3[1] = Mf[c * 16 + 13 + coff];
  const int r0 = (lane < 16) ? 0 : 2; // lane-half row split of B operand
  const float* chrow = x + (size_t)ch * TIN;
  v2f zB2, zB3; zB2[0] = zB2[1] = zB3[0] = zB3[1] = 0.0f;

  for (int blk = 0; blk < NBLK; ++blk) {
    const int t0 = p * LP + blk * 8;
    v2f xB0, xB1;
    if (t0 >= EDGE && t0 + 8 <= EDGE + TIN) {   // wave-uniform interior fast path
      const float* rp = chrow + (t0 - EDGE);
      xB0[0] = rp[r0];     xB0[1] = rp[r0 + 1];
      xB1[0] = rp[r0 + 4]; xB1[1] = rp[r0 + 5];
    } else {
      xB0[0] = load_ext(x, ch, t0 + r0);     xB0[1] = load_ext(x, ch, t0 + r0 + 1);
      xB1[0] = load_ext(x, ch, t0 + 4 + r0); xB1[1] = load_ext(x, ch, t0 + 4 + r0 + 1);
    }
    v8f d = {};
    d = wmma4(A0, xB0, d);     // x parts first: off the serial z chain
    d = wmma4(A1, xB1, d);
    d = wmma4(A2, zB2, d);
    d = wmma4(A3, zB3, d);
    if (lane < 16) {           // rows 0..7 = y outputs for this channel
      float* yp = yb + (size_t)t0 * NCH + ch;
      #pragma unroll
      for (int r = 0; r < 8; ++r) yp[(size_t)r * NCH] = d[r];
    }
    // new state (rows 8..15 live in lanes 16..31) -> B-operand form
    float s0 = __shfl(d[0], lane | 16, 32);
    float s1 = __shfl(d[1], lane | 16, 32);
    float s4 = __shfl(d[4], lane | 16, 32);
    float s5 = __shfl(d[5], lane | 16, 32);
    if (lane < 16) { zB2[0] = s0;   zB2[1] = s1;   zB3[0] = s4;   zB3[1] = s5;  }
    else           { zB2[0] = d[2]; zB2[1] = d[3]; zB3[0] = d[6]; zB3[1] = d[7]; }
    if (blk == NBLK - 1 && lane >= 16) {
      float* sp = sb + ((size_t)p * NCH + ch) * 8;
      #pragma unroll
      for (int i = 0; i < 8; ++i) sp[i] = d[i];
    }
  }
}

// ---------------- Phase B: compose partition entry states (both dirs) ------
__global__ __launch_bounds__(512) void kPhaseB(float* __restrict__ ws,
                                               const float* __restrict__ x,
                                               int backward) {
  int ch = threadIdx.x;
  if (ch >= NCH) return;
  const float* R   = ws + OFF_R;
  const float* ziF = ws + OFF_ZI;
  const float* sb  = ws + OFF_S;
  float* zin = ws + OFF_ZIN;
  float x0;
  if (!backward) {
    const float* row = x + (size_t)ch * TIN;
    x0 = 2.0f * row[0] - row[EDGE];                 // ext[0]
  } else {
    x0 = ws[OFF_Y + (size_t)(TEXT - 1) * NCH + ch]; // y[-1] of forward pass
  }
  float z[8];
  #pragma unroll
  for (int i = 0; i < 8; ++i) z[i] = ziF[i] * x0;
  #pragma unroll
  for (int i = 0; i < 8; ++i) zin[(size_t)ch * 8 + i] = z[i];
  for (int p = 1; p < NPART; ++p) {
    const float* sp = sb + ((size_t)(p - 1) * NCH + ch) * 8;
    float zn[8];
    #pragma unroll
    for (int r = 0; r < 8; ++r) {
      float acc = sp[r];
      #pragma unroll
      for (int q = 0; q < 8; ++q) acc += R[r * 8 + q] * z[q];
      zn[r] = acc;
    }
    float* zp = zin + ((size_t)p * NCH + ch) * 8;
    #pragma unroll
    for (int i = 0; i < 8; ++i) { z[i] = zn[i]; zp[i] = zn[i]; }
  }
}

// ---------------- Phase C forward: y += G[t%LP] . z_in[p] ------------------
__global__ __launch_bounds__(256) void kPhaseCFwd(float* __restrict__ ws) {
  size_t idx = (size_t)blockIdx.x * 256 + threadIdx.x;
  if (idx >= (size_t)TP * NCH) return;
  int t  = (int)(idx >> 9);
  int ch = (int)(idx & 511);
  int p = t / LP, tl = t - p * LP;
  const float* g = ws + OFF_G + (size_t)tl * 8;
  const float* z = ws + OFF_ZIN + ((size_t)p * NCH + ch) * 8;
  float acc = ws[OFF_Y + idx];
  #pragma unroll
  for (int i = 0; i < 8; ++i) acc += g[i] * z[i];
  ws[OFF_Y + idx] = acc;
}

// ---------------- Phase A backward: reversed scan, decimated store ----------
__global__ __launch_bounds__(32) void kPhaseABwd(float* __restrict__ ws,
                                                 float* __restrict__ out) {
  const int lane = threadIdx.x;
  const int p  = blockIdx.x >> 5;
  const int cg = blockIdx.x & 31;
  const int c  = lane & 15;
  const int ch = cg * 16 + c;
  const float* yb = ws + OFF_Y;
  float* sb = ws + OFF_S;
  const float* Mf = ws + OFF_M;
  const int coff = (lane >> 4) * 2;
  v2f A0, A1, A2, A3;
  A0[0] = Mf[c * 16 +  0 + coff]; A0[1] = Mf[c * 16 +  1 + coff];
  A1[0] = Mf[c * 16 +  4 + coff]; A1[1] = Mf[c * 16 +  5 + coff];
  A2[0] = Mf[c * 16 +  8 + coff]; A2[1] = Mf[c * 16 +  9 + coff];
  A3[0] = Mf[c * 16 + 12 + coff]; A3[1] = Mf[c * 16 + 13 + coff];
  const int r0 = (lane < 16) ? 0 : 2;
  v2f zB2, zB3; zB2[0] = zB2[1] = zB3[0] = zB3[1] = 0.0f;

  for (int blk = 0; blk < NBLK; ++blk) {
    const int t0 = p * LP + blk * 8;   // backward time
    v2f xB0, xB1;
    if (t0 + 8 <= TEXT) {              // wave-uniform fast path: all in range
      const float* bp = yb + (size_t)(TEXT - 1 - t0) * NCH + ch;
      xB0[0] = bp[-(size_t)(r0)     * NCH]; xB0[1] = bp[-(size_t)(r0 + 1) * NCH];
      xB1[0] = bp[-(size_t)(r0 + 4) * NCH]; xB1[1] = bp[-(size_t)(r0 + 5) * NCH];
    } else {
      xB0[0] = load_rev(yb, ch, t0 + r0);     xB0[1] = load_rev(yb, ch, t0 + r0 + 1);
      xB1[0] = load_rev(yb, ch, t0 + 4 + r0); xB1[1] = load_rev(yb, ch, t0 + 4 + r0 + 1);
    }
    v8f d = {};
    d = wmma4(A0, xB0, d);
    d = wmma4(A1, xB1, d);
    d = wmma4(A2, zB2, d);
    d = wmma4(A3, zB3, d);
    // t0 % 8 == 0 and (TEXT-1-EDGE) % 4 == 2  =>  only rows r=2 and r=6 hit the
    // decimated output grid tf = EDGE + 4k.
    {
      int tf2 = TEXT - 1 - (t0 + 2);   // row 2 forward-time
      int tf6 = TEXT - 1 - (t0 + 6);   // row 6 forward-time
      if (lane < 16) {
        float* orow = out + (size_t)ch * (TIN / QDEC);
        int d42 = tf2 - EDGE;
        if (d42 >= 0 && d42 <= 4 * 16383) orow[d42 >> 2] = d[2];
        int d46 = tf6 - EDGE;
        if (d46 >= 0 && d46 <= 4 * 16383) orow[d46 >> 2] = d[6];
      }
    }
    float s0 = __shfl(d[0], lane | 16, 32);
    float s1 = __shfl(d[1], lane | 16, 32);
    float s4 = __shfl(d[4], lane | 16, 32);
    float s5 = __shfl(d[5], lane | 16, 32);
    if (lane < 16) { zB2[0] = s0;   zB2[1] = s1;   zB3[0] = s4;   zB3[1] = s5;  }
    else           { zB2[0] = d[2]; zB2[1] = d[3]; zB3[0] = d[6]; zB3[1] = d[7]; }
    if (blk == NBLK - 1 && lane >= 16) {
      float* sp = sb + ((size_t)p * NCH + ch) * 8;
      #pragma unroll
      for (int i = 0; i < 8; ++i) sp[i] = d[i];
    }
  }
}

// ---------------- Phase C backward: correct decimated outputs ---------------
__global__ __launch_bounds__(256) void kPhaseCBwd(const float* __restrict__ ws,
                                                  float* __restrict__ out) {
  size_t idx = (size_t)blockIdx.x * 256 + threadIdx.x;
  if (idx >= (size_t)NCH * (TIN / QDEC)) return;
  int m = (int)(idx / (TIN / QDEC));
  int k = (int)(idx - (size_t)m * (TIN / QDEC));
  int tf = EDGE + 4 * k;
  int tb = TEXT - 1 - tf;
  int p = tb / LP, tl = tb - p * LP;
  const float* g = ws + OFF_G + (size_t)tl * 8;
  const float* z = ws + OFF_ZIN + ((size_t)p * NCH + m) * 8;
  float acc = out[idx];
  #pragma unroll
  for (int i = 0; i < 8; ++i) acc += g[i] * z[i];
  out[idx] = acc;
}

// ---------------- host: cheby1 design + lfilter_zi (double) ----------------
static void designFilter(Coef& cf) {
  const int N = 8;
  const double rp = 0.05, Wn = 0.8 / QDEC;
  double eps = std::sqrt(std::pow(10.0, 0.1 * rp) - 1.0);
  double mu  = std::asinh(1.0 / eps) / N;
  std::complex<double> p[8];
  for (int k = 1; k <= N; ++k) {
    double th = M_PI * (2.0 * k - 1.0) / (2.0 * N);
    p[k - 1] = std::complex<double>(-std::sinh(mu) * std::sin(th),
                                     std::cosh(mu) * std::cos(th));
  }
  std::complex<double> pr(1.0, 0.0);
  for (int i = 0; i < N; ++i) pr *= -p[i];
  double g = pr.real() / std::sqrt(1.0 + eps * eps);  // N even
  const double fs = 2.0;
  double warped = 2.0 * fs * std::tan(M_PI * Wn / fs);
  for (int i = 0; i < N; ++i) p[i] *= warped;
  g *= std::pow(warped, (double)N);
  const double fs2 = 2.0 * fs;
  std::complex<double> pd[8], den(1.0, 0.0);
  for (int i = 0; i < N; ++i) { pd[i] = (fs2 + p[i]) / (fs2 - p[i]); den *= (fs2 - p[i]); }
  double gd = g * (std::complex<double>(1.0, 0.0) / den).real();
  static const double binom[9] = {1, 8, 28, 56, 70, 56, 28, 8, 1};
  for (int i = 0; i <= N; ++i) cf.b[i] = gd * binom[i];
  std::complex<double> c[9];
  c[0] = 1.0; for (int i = 1; i <= N; ++i) c[i] = 0.0;
  for (int i = 0; i < N; ++i)
    for (int j = i + 1; j >= 1; --j) c[j] -= pd[i] * c[j - 1];
  for (int i = 0; i <= N; ++i) cf.a[i] = c[i].real();
  // lfilter_zi: solve (I - comp^T) zi = b[1:] - a[1:]*b[0]
  double comp[8][8] = {{0}};
  for (int j = 0; j < 8; ++j) comp[0][j] = -cf.a[j + 1] / cf.a[0];
  for (int i = 1; i < 8; ++i) comp[i][i - 1] = 1.0;
  double A9[8][8], Bv[8];
  for (int i = 0; i < 8; ++i)
    for (int j = 0; j < 8; ++j) A9[i][j] = (i == j ? 1.0 : 0.0) - comp[j][i];
  for (int i = 0; i < 8; ++i) Bv[i] = cf.b[i + 1] - cf.a[i + 1] * cf.b[0];
  for (int col = 0; col < 8; ++col) {        // partial-pivot Gauss
    int piv = col;
    for (int r = col + 1; r < 8; ++r)
      if (std::fabs(A9[r][col]) > std::fabs(A9[piv][col])) piv = r;
    if (piv != col) {
      for (int j = 0; j < 8; ++j) { double t = A9[col][j]; A9[col][j] = A9[piv][j]; A9[piv][j] = t; }
      double t = Bv[col]; Bv[col] = Bv[piv]; Bv[piv] = t;
    }
    for (int r = col + 1; r < 8; ++r) {
      double f = A9[r][col] / A9[col][col];
      for (int j = col; j < 8; ++j) A9[r][j] -= f * A9[col][j];
      Bv[r] -= f * Bv[col];
    }
  }
  for (int r = 7; r >= 0; --r) {
    double acc = Bv[r];
    for (int j = r + 1; j < 8; ++j) acc -= A9[r][j] * cf.zi[j];
    cf.zi[r] = acc / A9[r][r];
  }
}

extern "C" void kernel_launch(void* const* d_in, const int* in_sizes, int n_in,
                              void* d_out, int out_size, void* d_ws, size_t ws_size,
                              hipStream_t stream) {
  (void)in_sizes; (void)n_in; (void)out_size; (void)ws_size;
  const float* x = (const float*)d_in[0];
  float* out = (float*)d_out;
  float* ws  = (float*)d_ws;

  Coef cf;
  designFilter(cf);

  kPrecomp<<<1, 32, 0, stream>>>(cf, ws);
  // forward filtfilt pass
  kPhaseAFwd<<<NPART * 32, 32, 0, stream>>>(x, ws);
  kPhaseB<<<1, 512, 0, stream>>>(ws, x, 0);
  {
    size_t n = (size_t)TP * NCH;
    kPhaseCFwd<<<(unsigned)((n + 255) / 256), 256, 0, stream>>>(ws);
  }
  // backward filtfilt pass (reads corrected forward output reversed)
  kPhaseABwd<<<NPART * 32, 32, 0, stream>>>(ws, out);
  kPhaseB<<<1, 512, 0, stream>>>(ws, x, 1);
  {
    size_t n = (size_t)NCH * (TIN / QDEC);
    kPhaseCBwd<<<(unsigned)((n + 255) / 256), 256, 0, stream>>>(ws, out);
  }
}